// inputConvolutionHeads_21277267984593
// MI455X (gfx1250) — compile-verified
//
#include <hip/hip_runtime.h>

typedef __attribute__((ext_vector_type(2))) float v2f;
typedef __attribute__((ext_vector_type(8))) float v8f;

#define CC    12        // max input channels
#define KW    3         // conv kernel width
#define OUTC  128       // output channels
#define TLEN  8192      // time length
#define NT    128       // t-span per block
#define CKD   36        // packed K dimension = CC*KW
#define NJ    9         // 36 / 4 WMMA k-chunks
#define NE    18        // CKD/2 pre-paired B rows
#define BROW  288       // floats per B row: 128 pairs + 16 pad pairs (bank-shift 32)

__global__ __launch_bounds__(256)
void conv_heads_wmma_f32(const float* __restrict__ x,
                         const float* __restrict__ w12, const float* __restrict__ w6,
                         const float* __restrict__ w4,  const float* __restrict__ w3,
                         const float* __restrict__ w2,  float* __restrict__ out)
{
    __shared__ float Wlds[OUTC * CKD];   // packed A matrix, 18 KB
    __shared__ float Blds[NE * BROW];    // pre-paired B matrix, ~20.3 KB

    const int b    = blockIdx.y;
    const int t0   = blockIdx.x * NT;
    const int tid  = threadIdx.x;
    const int lane = tid & 31;
    const int wave = tid >> 5;

    // ---- routing: count valid leads (x[b, c, 0] != -inf) ----
    int count = 0;
    #pragma unroll
    for (int c = 0; c < CC; ++c)
        count += (x[((size_t)b * CC + c) * TLEN] != -__builtin_inff()) ? 1 : 0;

    const float* wsel = nullptr; int csel = 0;
    if      (count == 12) { wsel = w12; csel = 12; }
    else if (count == 6)  { wsel = w6;  csel = 6;  }
    else if (count == 4)  { wsel = w4;  csel = 4;  }
    else if (count == 3)  { wsel = w3;  csel = 3;  }
    else if (count == 2)  { wsel = w2;  csel = 2;  }

    // ---- stage packed weights A[o][ck], zero-padded to 12 channels ----
    // thread -> (o, half); (u+18)/3 == u/3 + 6 so all /3, %3 fold to constants.
    {
        const int o    = tid & 127;
        const int half = tid >> 7;            // 0 or 1
        const int ckb  = half * 18;
        #pragma unroll
        for (int u = 0; u < 18; ++u) {
            const int c = (u / 3) + half * 6; // compile-time u/3 + runtime half*6
            const int k = u % 3;              // compile-time
            float v = 0.0f;
            if (wsel != nullptr && c < csel) v = wsel[(o * csel + c) * KW + k];
            Wlds[o * CKD + ckb + u] = v;
        }
    }

    // ---- stage pre-paired B matrix via ASYNC global->LDS gather (CDNA5 path) ----
    // Blds[e][t][comp] = B[2e+comp][t] = xz[c][t0 + t + k - 1], ck = 2e+comp.
    // Valid elements: GLOBAL_LOAD_ASYNC_TO_LDS_B32 per-lane gather (ASYNCcnt).
    // Halo edges / invalid(-inf) channels: ds_store of 0.
    {
        const int tB   = tid & 127;           // t within tile
        const int half = tid >> 7;            // comp (0/1)
        #pragma unroll
        for (int i = 0; i < NE; ++i) {        // rows ck = 2i + half
            const int c = half ? ((2 * i + 1) / KW) : ((2 * i) / KW); // const-select
            const int k = half ? ((2 * i + 1) % KW) : ((2 * i) % KW); // const-select
            const int gt = t0 + tB + k - 1;
            const int li = i * BROW + tB * 2 + half;
            if (c < count && gt >= 0 && gt < TLEN) {
                const unsigned ldsoff = (unsigned)(size_t)&Blds[li];
                const float*   gaddr  = x + ((size_t)b * CC + c) * TLEN + gt;
                asm volatile("global_load_async_to_lds_b32 %0, %1, off"
                             :: "v"(ldsoff), "v"(gaddr) : "memory");
            } else {
                Blds[li] = 0.0f;
            }
        }
    }
    // prefetch next t-tile of x into cache hierarchy (global_prefetch_b8)
    if (t0 + NT < TLEN && tid < CC)
        __builtin_prefetch(&x[((size_t)b * CC + tid) * TLEN + t0 + NT], 0, 1);

    asm volatile("s_wait_asynccnt 0x0" ::: "memory");
    __syncthreads();

    // ---- WMMA fragments ----
    // A 16x4 f32: lanes 0-15 -> M=lane, K regs 4j+0/4j+1; lanes 16-31 -> K 4j+2/4j+3.
    const int ob  = wave * 16;           // this wave's o-tile
    const int m   = lane & 15;           // row (A) / col (B/D) within tile
    const int khi = lane >> 4;           // half-wave select

    // A fragments: one row base, compile-time 16j-byte offsets, ds_load_b64 each.
    const float* Wrow = &Wlds[(ob + m) * CKD + khi * 2];
    v2f afr[NJ];
    #pragma unroll
    for (int j = 0; j < NJ; ++j)
        afr[j] = *(const v2f*)&Wrow[4 * j];

    // B fragments: e = 2j + khi, column tb+m -> ONE base VGPR, immediate offsets.
    const float* Brow = &Blds[khi * BROW + m * 2];

    // D 16x16 f32: VGPR g -> M = g (+8 for high half-wave), N = m.
    const int orow = ob + (khi << 3);
    float* obase = out + ((size_t)b * OUTC + orow) * TLEN + (size_t)t0 + m;

    #pragma unroll
    for (int s = 0; s < NT / 16; ++s) {
        const int tb = s * 16;
        v8f acc = {0.f, 0.f, 0.f, 0.f, 0.f, 0.f, 0.f, 0.f};
        #pragma unroll
        for (int j = 0; j < NJ; ++j) {
            const v2f bfr = *(const v2f*)&Brow[j * (2 * BROW) + tb * 2];
            acc = __builtin_amdgcn_wmma_f32_16x16x4_f32(
                /*neg_a=*/false, afr[j], /*neg_b=*/false, bfr,
                /*c_mod=*/(short)0, acc, /*reuse_a=*/false, /*reuse_b=*/false);
        }
        #pragma unroll
        for (int g = 0; g < 8; ++g) {
            const float v = acc[g];
            obase[(size_t)g * TLEN + tb] = v > 0.0f ? v : 0.0f;  // ReLU
        }
    }
}

extern "C" void kernel_launch(void* const* d_in, const int* in_sizes, int n_in,
                              void* d_out, int out_size, void* d_ws, size_t ws_size,
                              hipStream_t stream) {
    (void)in_sizes; (void)n_in; (void)d_ws; (void)ws_size; (void)out_size;
    const float* x   = (const float*)d_in[0];
    const float* w12 = (const float*)d_in[1];
    const float* w6  = (const float*)d_in[2];
    const float* w4  = (const float*)d_in[3];
    const float* w3  = (const float*)d_in[4];
    const float* w2  = (const float*)d_in[5];
    float* out = (float*)d_out;

    dim3 grid(TLEN / NT, 32);  // (64 t-tiles, B=32)
    dim3 block(256);           // 8 wave32s
    conv_heads_wmma_f32<<<grid, block, 0, stream>>>(x, w12, w6, w4, w3, w2, out);
}